// GATv2Conv_32487132627455
// MI455X (gfx1250) — compile-verified
//
#include <hip/hip_runtime.h>
#include <hip/hip_bf16.h>

#define NN   100000
#define EE   1000000
#define DIN  128
#define DOUT 64
#define DE   11
#define NEG_SLOPE 0.2f
#define ROWS (EE + NN)           // edges + self loops
#define ROW_TILES (NN / 16)      // 6250 exactly

typedef float v2f __attribute__((ext_vector_type(2)));
typedef float v8f __attribute__((ext_vector_type(8)));

#if defined(__has_builtin)
#if __has_builtin(__builtin_amdgcn_wmma_f32_16x16x4_f32)
#define HAVE_WMMA_F32 1
#endif
#endif

// ---- monotone float <-> uint encoding for atomicMax on floats -------------
__device__ __forceinline__ unsigned f2mono(float f) {
    unsigned u = __float_as_uint(f);
    return (u & 0x80000000u) ? ~u : (u | 0x80000000u);
}
__device__ __forceinline__ float mono2f(unsigned u) {
    unsigned v = (u & 0x80000000u) ? (u & 0x7FFFFFFFu) : ~u;
    return __uint_as_float(v);
}

// ---- K0: zero accumulators (must run every launch: determinism) -----------
__global__ void gat_init(float* out, float* deg, float* denom, unsigned* lmax,
                         float* loop_attr) {
    long i = (long)blockIdx.x * blockDim.x + threadIdx.x;
    if (i < (long)NN * DOUT) out[i] = 0.0f;
    if (i < NN) { deg[i] = 0.0f; denom[i] = 0.0f; lmax[i] = 0u; }
    if (i < (long)NN * DE) loop_attr[i] = 0.0f;
}

// ---- K1: per-edge degree + incoming edge_attr sum per destination ---------
__global__ void gat_edge_accum(const int* __restrict__ ei,
                               const float* __restrict__ eattr,
                               float* __restrict__ deg,
                               float* __restrict__ loop_attr) {
    long e = (long)blockIdx.x * blockDim.x + threadIdx.x;
    if (e >= EE) return;
    int dst = ei[EE + e];
    atomicAdd(&deg[dst], 1.0f);
    const float* a = eattr + e * DE;
    float* la = loop_attr + (size_t)dst * DE;
#pragma unroll
    for (int j = 0; j < DE; ++j) atomicAdd(&la[j], a[j]);
}

// ---- K2: loop_attr /= max(deg,1) ------------------------------------------
__global__ void gat_loop_norm(float* __restrict__ loop_attr,
                              const float* __restrict__ deg) {
    long i = (long)blockIdx.x * blockDim.x + threadIdx.x;
    if (i >= (long)NN * DE) return;
    loop_attr[i] /= fmaxf(deg[i / DE], 1.0f);
}

// ---- K3: xl = x@W_l, xr = x@W_r via V_WMMA_F32_16X16X4_F32 ----------------
// One wave computes a 16x64 output tile (4 accumulators), K=128 in 32 steps.
#if HAVE_WMMA_F32
__global__ void gat_gemm(const float* __restrict__ x,
                         const float* __restrict__ Wl,
                         const float* __restrict__ Wr,
                         float* __restrict__ xl, float* __restrict__ xr) {
    int wave = threadIdx.x >> 5;
    int lane = threadIdx.x & 31;
    long tile = (long)blockIdx.x * 8 + wave;
    if (tile >= ROW_TILES) return;
    const float* W = blockIdx.y ? Wr : Wl;
    float* out = blockIdx.y ? xr : xl;
    int row0 = (int)tile * 16;
    int lhalf = lane >> 4;        // 0: K pair {0,1} / rows M=v ; 1: {2,3} / M=v+8
    int lmod  = lane & 15;

    v8f acc0 = {}, acc1 = {}, acc2 = {}, acc3 = {};
    const float* arow = x + (size_t)(row0 + lmod) * DIN + 2 * lhalf;

#pragma unroll 4
    for (int k0 = 0; k0 < DIN; k0 += 4) {
        v2f a; a.x = arow[k0]; a.y = arow[k0 + 1];
        const float* brow = W + (size_t)(k0 + 2 * lhalf) * DOUT + lmod;
        v2f b0, b1, b2, b3;
        b0.x = brow[0];  b0.y = brow[DOUT + 0];
        b1.x = brow[16]; b1.y = brow[DOUT + 16];
        b2.x = brow[32]; b2.y = brow[DOUT + 32];
        b3.x = brow[48]; b3.y = brow[DOUT + 48];
        acc0 = __builtin_amdgcn_wmma_f32_16x16x4_f32(false, a, false, b0, (short)0, acc0, false, false);
        acc1 = __builtin_amdgcn_wmma_f32_16x16x4_f32(false, a, false, b1, (short)0, acc1, false, false);
        acc2 = __builtin_amdgcn_wmma_f32_16x16x4_f32(false, a, false, b2, (short)0, acc2, false, false);
        acc3 = __builtin_amdgcn_wmma_f32_16x16x4_f32(false, a, false, b3, (short)0, acc3, false, false);
    }
#pragma unroll
    for (int v = 0; v < 8; ++v) {
        size_t r = (size_t)(row0 + v + 8 * lhalf) * DOUT + lmod;
        out[r +  0] = acc0[v];
        out[r + 16] = acc1[v];
        out[r + 32] = acc2[v];
        out[r + 48] = acc3[v];
    }
}
#else
// scalar fallback (same launch geometry): each lane computes 32 outputs
__global__ void gat_gemm(const float* __restrict__ x,
                         const float* __restrict__ Wl,
                         const float* __restrict__ Wr,
                         float* __restrict__ xl, float* __restrict__ xr) {
    int wave = threadIdx.x >> 5;
    int lane = threadIdx.x & 31;
    long tile = (long)blockIdx.x * 8 + wave;
    if (tile >= ROW_TILES) return;
    const float* W = blockIdx.y ? Wr : Wl;
    float* out = blockIdx.y ? xr : xl;
    int row0 = (int)tile * 16;
    for (int t = lane; t < 16 * DOUT; t += 32) {
        int r = t / DOUT, c = t % DOUT;
        float s = 0.0f;
        for (int k = 0; k < DIN; ++k)
            s = fmaf(x[(size_t)(row0 + r) * DIN + k], W[(size_t)k * DOUT + c], s);
        out[(size_t)(row0 + r) * DOUT + c] = s;
    }
}
#endif

// ---- K4: fused logits: leaky(xl[s]+xr[d]+attr@W_e) . att, segment max -----
// One wave per row (edge or self loop); lanes cover 2 output dims each.
__global__ void gat_logits(const int* __restrict__ ei,
                           const float* __restrict__ eattr,
                           const float* __restrict__ loop_attr,
                           const float* __restrict__ xl,
                           const float* __restrict__ xr,
                           const float* __restrict__ We,
                           const float* __restrict__ att,
                           float* __restrict__ logits,
                           unsigned* __restrict__ lmax) {
    int lane = threadIdx.x & 31;
    long f = ((long)blockIdx.x * blockDim.x + threadIdx.x) >> 5;
    if (f >= ROWS) return;
    int src, dst; const float* ap;
    if (f < EE) { src = ei[f]; dst = ei[EE + f]; ap = eattr + f * DE; }
    else        { src = dst = (int)(f - EE);     ap = loop_attr + (size_t)src * DE; }

    int k = lane * 2;
    float m0 = xl[(size_t)src * DOUT + k]     + xr[(size_t)dst * DOUT + k];
    float m1 = xl[(size_t)src * DOUT + k + 1] + xr[(size_t)dst * DOUT + k + 1];

    float av = (lane < DE) ? ap[lane] : 0.0f;
#pragma unroll
    for (int j = 0; j < DE; ++j) {
        float aj = __shfl(av, j, 32);
        m0 = fmaf(aj, We[j * DOUT + k],     m0);
        m1 = fmaf(aj, We[j * DOUT + k + 1], m1);
    }
    m0 = (m0 >= 0.0f) ? m0 : NEG_SLOPE * m0;
    m1 = (m1 >= 0.0f) ? m1 : NEG_SLOPE * m1;
    float p = fmaf(m0, att[k], m1 * att[k + 1]);
#pragma unroll
    for (int off = 16; off; off >>= 1) p += __shfl_down(p, off, 32);
    if (lane == 0) {
        logits[f] = p;
        atomicMax(&lmax[dst], f2mono(p));
    }
}

// ---- K5: ex = exp(logit - lmax[dst]); denom[dst] += ex (in-place) ---------
__global__ void gat_exp(const int* __restrict__ ei, float* __restrict__ logits,
                        const unsigned* __restrict__ lmax,
                        float* __restrict__ denom) {
    long f = (long)blockIdx.x * blockDim.x + threadIdx.x;
    if (f >= ROWS) return;
    int dst = (f < EE) ? ei[EE + f] : (int)(f - EE);
    float ex = __expf(logits[f] - mono2f(lmax[dst]));
    logits[f] = ex;                 // reuse buffer as ex
    atomicAdd(&denom[dst], ex);
}

// ---- K6: out[dst] += (ex/denom[dst]) * xl[src]; wave per row --------------
__global__ void gat_scatter(const int* __restrict__ ei,
                            const float* __restrict__ ex,
                            const float* __restrict__ denom,
                            const float* __restrict__ xl,
                            float* __restrict__ out) {
    int lane = threadIdx.x & 31;
    long f = ((long)blockIdx.x * blockDim.x + threadIdx.x) >> 5;
    if (f >= ROWS) return;
    int src, dst;
    if (f < EE) { src = ei[f]; dst = ei[EE + f]; }
    else        { src = dst = (int)(f - EE); }
    float alpha = ex[f] / denom[dst];
    int k = lane * 2;
    atomicAdd(&out[(size_t)dst * DOUT + k],     alpha * xl[(size_t)src * DOUT + k]);
    atomicAdd(&out[(size_t)dst * DOUT + k + 1], alpha * xl[(size_t)src * DOUT + k + 1]);
}

extern "C" void kernel_launch(void* const* d_in, const int* in_sizes, int n_in,
                              void* d_out, int out_size, void* d_ws, size_t ws_size,
                              hipStream_t stream) {
    const float* x     = (const float*)d_in[0];
    const int*   ei    = (const int*)  d_in[1];
    const float* eattr = (const float*)d_in[2];
    const float* Wl    = (const float*)d_in[3];
    const float* Wr    = (const float*)d_in[4];
    const float* We    = (const float*)d_in[5];
    const float* att   = (const float*)d_in[6];
    float* out = (float*)d_out;

    // workspace layout (256B-aligned chunks)
    char* base = (char*)d_ws;
    size_t off = 0;
    auto take = [&](size_t bytes) {
        char* p = base + off;
        off += (bytes + 255) & ~(size_t)255;
        return p;
    };
    float*    xl        = (float*)   take((size_t)NN * DOUT * 4);
    float*    xr        = (float*)   take((size_t)NN * DOUT * 4);
    float*    loop_attr = (float*)   take((size_t)NN * DE * 4);
    float*    deg       = (float*)   take((size_t)NN * 4);
    float*    denom     = (float*)   take((size_t)NN * 4);
    unsigned* lmax      = (unsigned*)take((size_t)NN * 4);
    float*    logits    = (float*)   take((size_t)ROWS * 4);  // reused as ex

    const int B = 256;
    // K0: zero accumulators + output
    gat_init<<<((long)NN * DOUT + B - 1) / B, B, 0, stream>>>(out, deg, denom, lmax, loop_attr);
    // K1: degree + loop_attr accumulation
    gat_edge_accum<<<(EE + B - 1) / B, B, 0, stream>>>(ei, eattr, deg, loop_attr);
    // K2: normalize loop_attr
    gat_loop_norm<<<((long)NN * DE + B - 1) / B, B, 0, stream>>>(loop_attr, deg);
    // K3: WMMA GEMMs (xl, xr); 8 waves/block, one 16x64 tile per wave
    dim3 ggrid((ROW_TILES + 7) / 8, 2, 1);
    gat_gemm<<<ggrid, B, 0, stream>>>(x, Wl, Wr, xl, xr);
    // K4: fused logits + segment max (wave per row)
    long waves = ROWS;
    gat_logits<<<(waves * 32 + B - 1) / B, B, 0, stream>>>(ei, eattr, loop_attr, xl, xr, We, att, logits, lmax);
    // K5: exp + segment sum
    gat_exp<<<(ROWS + B - 1) / B, B, 0, stream>>>(ei, logits, lmax, denom);
    // K6: alpha-weighted scatter (wave per row)
    gat_scatter<<<(waves * 32 + B - 1) / B, B, 0, stream>>>(ei, logits, denom, xl, out);
}